// PointNetFeaturePropagation__37245956391389
// MI455X (gfx1250) — compile-verified
//
#include <hip/hip_runtime.h>
#include <hip/hip_bf16.h>
#include <stddef.h>

// ---------------------------------------------------------------------------
// PointNet Feature Propagation for MI455X (gfx1250, wave32).
//
// Roofline: GEMMs total ~19.3 GFLOP; traffic ~350MB (~15us @ 23.3 TB/s) and
// the intermediate set is L2-resident (192MB).  Memory-bound => keep fp32 and
// use V_WMMA_F32_16X16X4_F32.  Each wave owns a 32(M)x64(N) tile so every
// LDS B-fragment feeds two wmmas (8 wmma per 2 global + 4 ds loads).
// ---------------------------------------------------------------------------

typedef __attribute__((ext_vector_type(2))) float v2f;
typedef __attribute__((ext_vector_type(8))) float v8f;

static constexpr int B    = 8;
static constexpr int N    = 8192;
static constexpr int S    = 2048;
static constexpr int D1   = 64;
static constexpr int D2   = 256;
static constexpr int CIN  = 320;          // D1 + D2
static constexpr int M0   = 256;
static constexpr int ROWS = B * N;        // 65536
static constexpr float BN_EPS = 1e-5f;

static constexpr int LDSW_STRIDE = 324;   // >= K+4; even (8B align); %64==4 (bank spread)

// ---------------------------------------------------------------------------
// 0) points2 (B,D2,S) -> p2t (B,S,D2), 32x32 LDS-tiled transpose
// ---------------------------------------------------------------------------
__global__ __launch_bounds__(256) void transpose_p2(const float* __restrict__ p2,
                                                    float* __restrict__ p2t) {
  __shared__ float tile[32][33];
  const int b  = blockIdx.z;
  const int s0 = blockIdx.x * 32;
  const int c0 = blockIdx.y * 32;
  const int tx = threadIdx.x, ty = threadIdx.y;  // (32,8)
  const float* src = p2 + (size_t)b * D2 * S;
#pragma unroll
  for (int j = 0; j < 32; j += 8)
    tile[ty + j][tx] = src[(size_t)(c0 + ty + j) * S + s0 + tx];
  __syncthreads();
  float* dst = p2t + (size_t)b * S * D2;
#pragma unroll
  for (int j = 0; j < 32; j += 8)
    dst[(size_t)(s0 + ty + j) * D2 + c0 + tx] = tile[tx][ty + j];
}

// ---------------------------------------------------------------------------
// 1) 3-NN search: one thread per query point; xyz2 staged in LDS (24KB)
// ---------------------------------------------------------------------------
__global__ __launch_bounds__(256) void knn3(const float* __restrict__ xyz1,
                                            const float* __restrict__ xyz2,
                                            int* __restrict__ idxOut,
                                            float* __restrict__ wOut) {
  __shared__ float sx[S], sy[S], sz[S];
  const int b = blockIdx.y;
  const int n = blockIdx.x * blockDim.x + threadIdx.x;
  const float* x2 = xyz2 + (size_t)b * 3 * S;
  for (int i = threadIdx.x; i < S; i += blockDim.x) {
    sx[i] = x2[i];
    sy[i] = x2[S + i];
    sz[i] = x2[2 * S + i];
  }
  __syncthreads();

  const float* x1 = xyz1 + (size_t)b * 3 * N;
  const float px = x1[n], py = x1[N + n], pz = x1[2 * N + n];

  float d0 = 3.4e38f, d1 = 3.4e38f, d2 = 3.4e38f;
  int   i0 = 0, i1 = 0, i2 = 0;
  for (int s = 0; s < S; ++s) {
    const float dx = px - sx[s], dy = py - sy[s], dz = pz - sz[s];
    const float d = dx * dx + dy * dy + dz * dz;
    if (d < d0)      { d2 = d1; i2 = i1; d1 = d0; i1 = i0; d0 = d; i0 = s; }
    else if (d < d1) { d2 = d1; i2 = i1; d1 = d;  i1 = s; }
    else if (d < d2) { d2 = d;  i2 = s; }
  }
  const float r0 = 1.f / (d0 + 1e-8f);
  const float r1 = 1.f / (d1 + 1e-8f);
  const float r2 = 1.f / (d2 + 1e-8f);
  const float rs = 1.f / (r0 + r1 + r2);
  const size_t r = (size_t)b * N + n;
  idxOut[r * 3 + 0] = i0; idxOut[r * 3 + 1] = i1; idxOut[r * 3 + 2] = i2;
  wOut[r * 3 + 0] = r0 * rs; wOut[r * 3 + 1] = r1 * rs; wOut[r * 3 + 2] = r2 * rs;
}

// ---------------------------------------------------------------------------
// 2) interpolate + concat -> A0 (ROWS x CIN) row-major
// ---------------------------------------------------------------------------
__global__ __launch_bounds__(256) void interp_concat(const float* __restrict__ p2t,
                                                     const float* __restrict__ points1,
                                                     const int* __restrict__ idx3,
                                                     const float* __restrict__ w3,
                                                     float* __restrict__ A0) {
  const size_t r = blockIdx.x;          // b*N + n
  const int b = (int)(r >> 13);         // / 8192
  const int n = (int)(r & (N - 1));
  const int c = threadIdx.x;            // 0..255
  const int i0 = idx3[r * 3 + 0], i1 = idx3[r * 3 + 1], i2 = idx3[r * 3 + 2];
  const float w0 = w3[r * 3 + 0], w1 = w3[r * 3 + 1], w2 = w3[r * 3 + 2];
  const float* base = p2t + (size_t)b * S * D2;
  const float v = w0 * base[(size_t)i0 * D2 + c] +
                  w1 * base[(size_t)i1 * D2 + c] +
                  w2 * base[(size_t)i2 * D2 + c];
  A0[r * CIN + D1 + c] = v;
  if (c < D1)
    A0[r * CIN + c] = points1[((size_t)b * D1 + c) * N + n];
}

// ---------------------------------------------------------------------------
// 3) GEMM + bias: Y (ROWS x OC) = A (ROWS x K) * W(OC x K)^T + bias
//    V_WMMA_F32_16X16X4_F32.  Block = 8 waves; each wave -> 32(M) x 64(N)
//    (2 M-tiles x 4 N-tiles = 8 accumulators), weight strip in LDS.
// ---------------------------------------------------------------------------
__global__ __launch_bounds__(256) void gemm_bias_wmma(const float* __restrict__ A,
                                                      const float* __restrict__ W,
                                                      const float* __restrict__ bias,
                                                      float* __restrict__ Y,
                                                      int K, int OC) {
  extern __shared__ float ldsW[];  // 64 * LDSW_STRIDE floats
  const int tid   = threadIdx.x;
  const int wave  = tid >> 5;
  const int lane  = tid & 31;
  const int nbase = blockIdx.y * 64;
  const int mbase = blockIdx.x * 256 + wave * 32;

  // Stage weight strip: rows nbase..nbase+63, all K columns (float4, coalesced)
  {
    const int orow = tid >> 2;                 // 0..63
    for (int k = (tid & 3) * 4; k < K; k += 16) {
      const float4 w4 = *(const float4*)&W[(size_t)(nbase + orow) * K + k];
      float* dst = &ldsW[orow * LDSW_STRIDE + k];
      dst[0] = w4.x; dst[1] = w4.y; dst[2] = w4.z; dst[3] = w4.w;
    }
  }
  __syncthreads();

  v8f acc0[4] = {};
  v8f acc1[4] = {};
  const int l15   = lane & 15;
  const int khalf = (lane >> 4) * 2;           // 0 (lanes 0-15) or 2 (lanes 16-31)
  const float* aptr0 = A + (size_t)(mbase + l15) * K + khalf;
  const float* aptr1 = aptr0 + (size_t)16 * K;

#pragma unroll 4
  for (int k = 0; k < K; k += 4) {
    const v2f a0 = *(const v2f*)(aptr0 + k);
    const v2f a1 = *(const v2f*)(aptr1 + k);
    v2f bf[4];
#pragma unroll
    for (int t = 0; t < 4; ++t)
      bf[t] = *(const v2f*)&ldsW[(t * 16 + l15) * LDSW_STRIDE + k + khalf];
#pragma unroll
    for (int t = 0; t < 4; ++t) {
      acc0[t] = __builtin_amdgcn_wmma_f32_16x16x4_f32(
          false, a0, false, bf[t], (short)0, acc0[t], false, false);
      acc1[t] = __builtin_amdgcn_wmma_f32_16x16x4_f32(
          false, a1, false, bf[t], (short)0, acc1[t], false, false);
    }
  }

  // C/D layout: VGPR e -> M = tilebase + e + 8*(lane>=16); N = nbase + t*16 + (lane&15)
  const int mrow0 = mbase + (lane >> 4) * 8;
#pragma unroll
  for (int t = 0; t < 4; ++t) {
    const int o = nbase + t * 16 + l15;
    const float bv = bias[o];
#pragma unroll
    for (int e = 0; e < 8; ++e) {
      Y[(size_t)(mrow0 + e) * OC + o]      = acc0[t][e] + bv;
      Y[(size_t)(mrow0 + 16 + e) * OC + o] = acc1[t][e] + bv;
    }
  }
}

// ---------------------------------------------------------------------------
// 4) BatchNorm statistics: deterministic two-stage column reduction
// ---------------------------------------------------------------------------
__global__ __launch_bounds__(256) void col_partial_stats(const float* __restrict__ Y,
                                                         float* __restrict__ part,
                                                         int rowsPerBlock) {
  const int c  = threadIdx.x;                  // channel 0..255
  const size_t r0 = (size_t)blockIdx.x * rowsPerBlock;
  float s = 0.f, ss = 0.f;
  for (int r = 0; r < rowsPerBlock; ++r) {
    const float v = Y[(r0 + r) * M0 + c];
    s += v; ss += v * v;
  }
  part[(size_t)blockIdx.x * M0 + c] = s;
  part[(size_t)gridDim.x * M0 + (size_t)blockIdx.x * M0 + c] = ss;
}

__global__ __launch_bounds__(256) void finalize_stats(const float* __restrict__ part,
                                                      const float* __restrict__ g,
                                                      const float* __restrict__ be,
                                                      float* __restrict__ scale,
                                                      float* __restrict__ shift,
                                                      int nparts, float invCount) {
  const int c = threadIdx.x;
  float s = 0.f, ss = 0.f;
  for (int p = 0; p < nparts; ++p) {
    s  += part[(size_t)p * M0 + c];
    ss += part[(size_t)nparts * M0 + (size_t)p * M0 + c];
  }
  const float mu  = s * invCount;
  const float var = ss * invCount - mu * mu;
  const float sc  = g[c] * rsqrtf(var + BN_EPS);
  scale[c] = sc;
  shift[c] = be[c] - mu * sc;
}

__global__ __launch_bounds__(256) void bn_relu_inplace(float* __restrict__ Y,
                                                       const float* __restrict__ scale,
                                                       const float* __restrict__ shift) {
  const size_t i = (size_t)blockIdx.x * blockDim.x + threadIdx.x;
  const int c = (int)(i & (M0 - 1));
  const float v = Y[i] * scale[c] + shift[c];
  Y[i] = v > 0.f ? v : 0.f;
}

// ---------------------------------------------------------------------------
// 5) BN + ReLU fused with the (ROWS x C) -> (B, C, N) output transpose
// ---------------------------------------------------------------------------
__global__ __launch_bounds__(256) void bn_relu_transpose_out(const float* __restrict__ Y,
                                                             const float* __restrict__ scale,
                                                             const float* __restrict__ shift,
                                                             float* __restrict__ out) {
  __shared__ float tile[32][33];
  const int b  = blockIdx.z;
  const int n0 = blockIdx.x * 32;
  const int c0 = blockIdx.y * 32;
  const int tx = threadIdx.x, ty = threadIdx.y;  // (32,8)
  const float* src = Y + (size_t)b * N * M0;
#pragma unroll
  for (int j = 0; j < 32; j += 8)
    tile[ty + j][tx] = src[(size_t)(n0 + ty + j) * M0 + c0 + tx];
  __syncthreads();
  float* dst = out + (size_t)b * M0 * N;
#pragma unroll
  for (int j = 0; j < 32; j += 8) {
    const int c = c0 + ty + j;
    const float v = tile[tx][ty + j] * scale[c] + shift[c];
    dst[(size_t)c * N + n0 + tx] = v > 0.f ? v : 0.f;
  }
}

// ---------------------------------------------------------------------------
// Launch
// ---------------------------------------------------------------------------
extern "C" void kernel_launch(void* const* d_in, const int* in_sizes, int n_in,
                              void* d_out, int out_size, void* d_ws, size_t ws_size,
                              hipStream_t stream) {
  const float* xyz1    = (const float*)d_in[0];
  const float* xyz2    = (const float*)d_in[1];
  const float* points1 = (const float*)d_in[2];
  const float* points2 = (const float*)d_in[3];
  const float* w0 = (const float*)d_in[4];
  const float* b0 = (const float*)d_in[5];
  const float* g0 = (const float*)d_in[6];
  const float* be0 = (const float*)d_in[7];
  const float* w1 = (const float*)d_in[8];
  const float* b1 = (const float*)d_in[9];
  const float* g1 = (const float*)d_in[10];
  const float* be1 = (const float*)d_in[11];
  float* out = (float*)d_out;

  // Workspace layout (floats).  p2t/idx/w alias the Y0 region: they are fully
  // consumed by interp_concat before GEMM0 writes Y0.  Y1 aliases A0.
  float* ws = (float*)d_ws;
  float* A0  = ws;                                   // ROWS*CIN  = 20,971,520 f
  float* Y0  = ws + (size_t)ROWS * CIN;              // ROWS*M0   = 16,777,216 f
  float* p2t = Y0;                                   // B*S*D2    =  4,194,304 f
  int*   idx3 = (int*)(Y0 + (size_t)B * S * D2);     // ROWS*3
  float* wgt3 = Y0 + (size_t)B * S * D2 + (size_t)ROWS * 3;
  float* stats = Y0 + (size_t)ROWS * M0;             // partials + scale + shift
  float* partials = stats;                           // 2*64*256 f
  float* scaleBuf = stats + 2 * 64 * M0;             // 256 f
  float* shiftBuf = scaleBuf + M0;                   // 256 f
  float* Y1 = A0;                                    // reuse A0 for layer-1 output

  const int gemmLds = 64 * LDSW_STRIDE * (int)sizeof(float);  // 82,944 B

  // 0) transpose points2
  transpose_p2<<<dim3(S / 32, D2 / 32, B), dim3(32, 8), 0, stream>>>(points2, p2t);
  // 1) 3-NN
  knn3<<<dim3(N / 256, B), 256, 0, stream>>>(xyz1, xyz2, idx3, wgt3);
  // 2) interpolate + concat -> A0
  interp_concat<<<dim3(ROWS), 256, 0, stream>>>(p2t, points1, idx3, wgt3, A0);
  // 3) layer 0 GEMM (K=320)
  gemm_bias_wmma<<<dim3(ROWS / 256, M0 / 64), 256, gemmLds, stream>>>(A0, w0, b0, Y0, CIN, M0);
  // 4) layer 0 BN stats + apply
  col_partial_stats<<<dim3(64), 256, 0, stream>>>(Y0, partials, ROWS / 64);
  finalize_stats<<<dim3(1), 256, 0, stream>>>(partials, g0, be0, scaleBuf, shiftBuf,
                                              64, 1.0f / (float)ROWS);
  bn_relu_inplace<<<dim3((ROWS * M0) / 256), 256, 0, stream>>>(Y0, scaleBuf, shiftBuf);
  // 5) layer 1 GEMM (K=256), output into Y1 (= A0 region)
  gemm_bias_wmma<<<dim3(ROWS / 256, M0 / 64), 256, gemmLds, stream>>>(Y0, w1, b1, Y1, M0, M0);
  // 6) layer 1 BN stats
  col_partial_stats<<<dim3(64), 256, 0, stream>>>(Y1, partials, ROWS / 64);
  finalize_stats<<<dim3(1), 256, 0, stream>>>(partials, g1, be1, scaleBuf, shiftBuf,
                                              64, 1.0f / (float)ROWS);
  // 7) BN + ReLU fused with output transpose to (B, 256, N)
  bn_relu_transpose_out<<<dim3(N / 32, M0 / 32, B), dim3(32, 8), 0, stream>>>(
      Y1, scaleBuf, shiftBuf, out);
}